// GATModel_29411936043505
// MI455X (gfx1250) — compile-verified
//
#include <hip/hip_runtime.h>
#include <math.h>

#define N_NODES   50000
#define N_EDGES   800000
#define E_TOT     (N_EDGES + N_NODES)   /* 850000 incl. self loops */
#define N_GRAPHS  512
#define FEAT      256
#define HEADS     4
#define HDIM      64
#define NEG_SLOPE 0.2f

typedef float v2f __attribute__((ext_vector_type(2)));
typedef float v8f __attribute__((ext_vector_type(8)));

static __device__ __forceinline__ float leaky(float x) {
  return x > 0.f ? x : NEG_SLOPE * x;
}
static __device__ __forceinline__ void edge_sd(const int* __restrict__ ei, int e,
                                               int& s, int& d) {
  if (e < N_EDGES) { s = ei[e]; d = ei[N_EDGES + e]; }
  else             { s = d = e - N_EDGES; }
}

// ---------------------------------------------------------------------------
// GEMM: C[M,256] = A[M,K] @ W[K,256], f32 via V_WMMA_F32_16X16X4_F32.
// Wave = 16(M) x 64(N) strip; block(256) = 8 waves = 32(M) x 256(N) tile.
// ---------------------------------------------------------------------------
__global__ __launch_bounds__(256) void gemm_wmma_f32(
    const float* __restrict__ A, const float* __restrict__ W,
    float* __restrict__ C, int M, int K) {
  const int Nn   = FEAT;
  const int lane = threadIdx.x & 31;
  const int wave = threadIdx.x >> 5;
  const int mbase = blockIdx.x * 32 + (wave >> 2) * 16;
  const int nbase = (wave & 3) * 64;
  const int half  = lane >> 4;
  const int l16   = lane & 15;
  int arow = mbase + l16; if (arow >= M) arow = M - 1;   // clamp (tail rows unused)
  const float* arp  = A + (size_t)arow * K;
  const int    koff = half * 2;

  v8f acc0 = {}, acc1 = {}, acc2 = {}, acc3 = {};
  for (int k = 0; k < K; k += 4) {
    v2f a = *(const v2f*)(arp + k + koff);               // contiguous pair
    const float* w0 = W + (size_t)(k + koff) * Nn + nbase + l16;
    const float* w1 = w0 + Nn;
    v2f b0 = { w0[0],  w1[0]  };
    v2f b1 = { w0[16], w1[16] };
    v2f b2 = { w0[32], w1[32] };
    v2f b3 = { w0[48], w1[48] };
    acc0 = __builtin_amdgcn_wmma_f32_16x16x4_f32(false, a, false, b0, (short)0, acc0, false, false);
    acc1 = __builtin_amdgcn_wmma_f32_16x16x4_f32(false, a, false, b1, (short)0, acc1, false, false);
    acc2 = __builtin_amdgcn_wmma_f32_16x16x4_f32(false, a, false, b2, (short)0, acc2, false, false);
    acc3 = __builtin_amdgcn_wmma_f32_16x16x4_f32(false, a, false, b3, (short)0, acc3, false, false);
  }
#pragma unroll
  for (int r = 0; r < 8; ++r) {
    int row = mbase + r + half * 8;
    if (row < M) {
      float* cp = C + (size_t)row * Nn + nbase + l16;
      cp[0]  = acc0[r];
      cp[16] = acc1[r];
      cp[32] = acc2[r];
      cp[48] = acc3[r];
    }
  }
}

// alpha_s[n,h] = <h[n,h,:], a_src[h,:]>, alpha_d likewise
__global__ void attn_coef(const float* __restrict__ h,
                          const float* __restrict__ asrc,
                          const float* __restrict__ adst,
                          float* __restrict__ as_, float* __restrict__ ad_) {
  int gid = blockIdx.x * blockDim.x + threadIdx.x;
  if (gid >= N_NODES * HEADS) return;
  int n = gid >> 2, hd = gid & 3;
  const float* hp = h + (size_t)n * FEAT + hd * HDIM;
  const float* ap = asrc + hd * HDIM;
  const float* dp = adst + hd * HDIM;
  float s = 0.f, d = 0.f;
#pragma unroll 8
  for (int i = 0; i < HDIM; ++i) { float v = hp[i]; s += v * ap[i]; d += v * dp[i]; }
  as_[gid] = s; ad_[gid] = d;
}

// ------------------------- CSR build (once per launch) ----------------------
__global__ void csr_zero(int* __restrict__ deg) {
  int gid = blockIdx.x * blockDim.x + threadIdx.x;
  if (gid < N_NODES) deg[gid] = 0;
}

__global__ void csr_count(const int* __restrict__ ei, int* __restrict__ deg) {
  int e = blockIdx.x * blockDim.x + threadIdx.x;
  if (e >= E_TOT) return;
  int s, d; edge_sd(ei, e, s, d);
  atomicAdd(&deg[d], 1);
}

// single-block exclusive scan: deg -> row_start[0..N], cursor copy
__global__ __launch_bounds__(1024) void csr_scan(const int* __restrict__ deg,
                                                 int* __restrict__ row_start,
                                                 int* __restrict__ cursor) {
  __shared__ int part[1024];
  const int t  = threadIdx.x;
  const int CH = (N_NODES + 1023) / 1024;           // 49 nodes per thread
  int lo = t * CH, hi = lo + CH; if (hi > N_NODES) hi = N_NODES;
  int s = 0;
  for (int n = lo; n < hi; ++n) s += deg[n];
  part[t] = s;
  __syncthreads();
  for (int off = 1; off < 1024; off <<= 1) {        // Hillis-Steele inclusive scan
    int v = (t >= off) ? part[t - off] : 0;
    __syncthreads();
    part[t] += v;
    __syncthreads();
  }
  int base = (t == 0) ? 0 : part[t - 1];
  for (int n = lo; n < hi; ++n) { row_start[n] = base; cursor[n] = base; base += deg[n]; }
  if (t == 1023) row_start[N_NODES] = part[1023];   // == E_TOT
}

__global__ void csr_scatter(const int* __restrict__ ei, int* __restrict__ cursor,
                            int* __restrict__ csr_src) {
  int e = blockIdx.x * blockDim.x + threadIdx.x;
  if (e >= E_TOT) return;
  int s, d; edge_sd(ei, e, s, d);
  int pos = atomicAdd(&cursor[d], 1);
  csr_src[pos] = s;
}

// ---------------------------------------------------------------------------
// Fused, atomic-free segment softmax + aggregation. One wave32 per node:
//  pass 1: lane-strided max of leaky logits per head (butterfly reduce)
//  pass 2: lane-strided sum of exp(logit - m)      (butterfly reduce)
//  pass 3: sequential edge loop, each lane owns 8 channels (2x float4 gather)
// out[n,:] = sum_e alpha_e * h[src_e,:] + bias
// ---------------------------------------------------------------------------
__global__ __launch_bounds__(256) void node_gat_aggr(
    const float* __restrict__ h, const float* __restrict__ as_,
    const float* __restrict__ ad_, const int* __restrict__ row_start,
    const int* __restrict__ csr_src, const float* __restrict__ bias,
    float* __restrict__ out) {
  const int n    = (blockIdx.x * blockDim.x + threadIdx.x) >> 5;
  const int lane = threadIdx.x & 31;
  if (n >= N_NODES) return;
  const int beg = row_start[n], end = row_start[n + 1];

  const float ad0 = ad_[n * 4 + 0], ad1 = ad_[n * 4 + 1];
  const float ad2 = ad_[n * 4 + 2], ad3 = ad_[n * 4 + 3];

  // pass 1: per-head max
  float mx0 = -3.4e38f, mx1 = -3.4e38f, mx2 = -3.4e38f, mx3 = -3.4e38f;
  for (int i = beg + lane; i < end; i += 32) {
    const float* ap = as_ + (size_t)csr_src[i] * 4;
    mx0 = fmaxf(mx0, leaky(ap[0] + ad0));
    mx1 = fmaxf(mx1, leaky(ap[1] + ad1));
    mx2 = fmaxf(mx2, leaky(ap[2] + ad2));
    mx3 = fmaxf(mx3, leaky(ap[3] + ad3));
  }
#pragma unroll
  for (int m = 16; m > 0; m >>= 1) {
    mx0 = fmaxf(mx0, __shfl_xor(mx0, m, 32));
    mx1 = fmaxf(mx1, __shfl_xor(mx1, m, 32));
    mx2 = fmaxf(mx2, __shfl_xor(mx2, m, 32));
    mx3 = fmaxf(mx3, __shfl_xor(mx3, m, 32));
  }

  // pass 2: per-head denom
  float sm0 = 0.f, sm1 = 0.f, sm2 = 0.f, sm3 = 0.f;
  for (int i = beg + lane; i < end; i += 32) {
    const float* ap = as_ + (size_t)csr_src[i] * 4;
    sm0 += __expf(leaky(ap[0] + ad0) - mx0);
    sm1 += __expf(leaky(ap[1] + ad1) - mx1);
    sm2 += __expf(leaky(ap[2] + ad2) - mx2);
    sm3 += __expf(leaky(ap[3] + ad3) - mx3);
  }
#pragma unroll
  for (int m = 16; m > 0; m >>= 1) {
    sm0 += __shfl_xor(sm0, m, 32);
    sm1 += __shfl_xor(sm1, m, 32);
    sm2 += __shfl_xor(sm2, m, 32);
    sm3 += __shfl_xor(sm3, m, 32);
  }

  // pass 3: aggregation; lane owns channels [lane*8, lane*8+8) -> head = lane>>3
  const int   hd   = lane >> 3;
  const float mxh  = (hd == 0) ? mx0 : (hd == 1) ? mx1 : (hd == 2) ? mx2 : mx3;
  const float rsmh = 1.f / ((hd == 0) ? sm0 : (hd == 1) ? sm1 : (hd == 2) ? sm2 : sm3);
  const float adh  = ad_[n * 4 + hd];
  const int   c0   = lane * 8;
  float a0 = 0.f, a1 = 0.f, a2 = 0.f, a3 = 0.f, a4 = 0.f, a5 = 0.f, a6 = 0.f, a7 = 0.f;
  for (int i = beg; i < end; ++i) {
    int s = csr_src[i];                       // uniform across wave -> broadcast
    float alpha = __expf(leaky(as_[(size_t)s * 4 + hd] + adh) - mxh) * rsmh;
    const float4* hp = (const float4*)(h + (size_t)s * FEAT + c0);
    float4 u = hp[0], v = hp[1];
    a0 += u.x * alpha; a1 += u.y * alpha; a2 += u.z * alpha; a3 += u.w * alpha;
    a4 += v.x * alpha; a5 += v.y * alpha; a6 += v.z * alpha; a7 += v.w * alpha;
  }
  const float* bp = bias + c0;
  float* op = out + (size_t)n * FEAT + c0;
  op[0] = a0 + bp[0]; op[1] = a1 + bp[1]; op[2] = a2 + bp[2]; op[3] = a3 + bp[3];
  op[4] = a4 + bp[4]; op[5] = a5 + bp[5]; op[6] = a6 + bp[6]; op[7] = a7 + bp[7];
}

// ------------------------------ pooling + FC --------------------------------
__global__ void pool_zero(float* __restrict__ pooled) {
  int gid = blockIdx.x * blockDim.x + threadIdx.x;
  if (gid < N_GRAPHS * FEAT) pooled[gid] = 0.f;
}

__global__ void pool_sum(const float* __restrict__ h, const int* __restrict__ batch,
                         float* __restrict__ pooled) {
  int gid = blockIdx.x * blockDim.x + threadIdx.x;
  if (gid >= N_NODES * (FEAT / 4)) return;
  int n  = gid >> 6;
  int c0 = (gid & 63) * 4;
  int g  = batch[n];
  float4 hv = *(const float4*)(h + (size_t)n * FEAT + c0);
  float* pp = pooled + (size_t)g * FEAT + c0;
  atomicAdd(pp + 0, hv.x);
  atomicAdd(pp + 1, hv.y);
  atomicAdd(pp + 2, hv.z);
  atomicAdd(pp + 3, hv.w);
}

__global__ __launch_bounds__(256) void final_fc(const float* __restrict__ pooled,
                                                const float* __restrict__ fcW,
                                                const float* __restrict__ fcb,
                                                float* __restrict__ out) {
  __shared__ float red[256];
  int g = blockIdx.x, t = threadIdx.x;
  red[t] = pooled[(size_t)g * FEAT + t] * fcW[t];
  __syncthreads();
  for (int s = 128; s > 0; s >>= 1) {
    if (t < s) red[t] += red[t + s];
    __syncthreads();
  }
  if (t == 0) out[g] = red[0] + fcb[0];
}

static inline int cdiv(long long a, long long b) { return (int)((a + b - 1) / b); }

extern "C" void kernel_launch(void* const* d_in, const int* in_sizes, int n_in,
                              void* d_out, int out_size, void* d_ws, size_t ws_size,
                              hipStream_t stream) {
  const float* x     = (const float*)d_in[0];
  const int*   ei    = (const int*)  d_in[1];
  const int*   batch = (const int*)  d_in[2];
  const float* Wl[3]   = { (const float*)d_in[3],  (const float*)d_in[7],  (const float*)d_in[11] };
  const float* asrc[3] = { (const float*)d_in[4],  (const float*)d_in[8],  (const float*)d_in[12] };
  const float* adst[3] = { (const float*)d_in[5],  (const float*)d_in[9],  (const float*)d_in[13] };
  const float* bias[3] = { (const float*)d_in[6],  (const float*)d_in[10], (const float*)d_in[14] };
  const float* fcW = (const float*)d_in[15];
  const float* fcb = (const float*)d_in[16];
  float* out = (float*)d_out;

  // workspace carve-up (256B aligned)
  char* ws = (char*)d_ws;
  size_t off = 0;
  auto alloc = [&](size_t bytes) { char* p = ws + off; off = (off + bytes + 255) & ~(size_t)255; return p; };
  float* hb0       = (float*)alloc((size_t)N_NODES * FEAT * 4);
  float* hb1       = (float*)alloc((size_t)N_NODES * FEAT * 4);
  float* as_       = (float*)alloc((size_t)N_NODES * HEADS * 4);
  float* ad_       = (float*)alloc((size_t)N_NODES * HEADS * 4);
  float* pooled    = (float*)alloc((size_t)N_GRAPHS * FEAT * 4);
  int*   deg       = (int*)  alloc((size_t)N_NODES * 4);
  int*   row_start = (int*)  alloc((size_t)(N_NODES + 1) * 4);
  int*   cursor    = (int*)  alloc((size_t)N_NODES * 4);
  int*   csr_src   = (int*)  alloc((size_t)E_TOT * 4);
  (void)ws_size; (void)n_in; (void)in_sizes; (void)out_size;

  const int TB = 256;
  const int g_gemm = cdiv(N_NODES, 32);
  const int g_nh   = cdiv((long long)N_NODES * HEADS, TB);
  const int g_nf4  = cdiv((long long)N_NODES * (FEAT / 4), TB);
  const int g_e    = cdiv(E_TOT, TB);
  const int g_node = cdiv((long long)N_NODES * 32, TB);   // one wave per node

  // build dst-CSR (graph is identical across all three layers)
  csr_zero   <<<cdiv(N_NODES, TB), TB, 0, stream>>>(deg);
  csr_count  <<<g_e, TB, 0, stream>>>(ei, deg);
  csr_scan   <<<1, 1024, 0, stream>>>(deg, row_start, cursor);
  csr_scatter<<<g_e, TB, 0, stream>>>(ei, cursor, csr_src);

  const float* lin = x;     // GEMM input
  int K = 128;
  for (int l = 0; l < 3; ++l) {
    gemm_wmma_f32<<<g_gemm, TB, 0, stream>>>(lin, Wl[l], hb0, N_NODES, K);
    attn_coef    <<<g_nh,   TB, 0, stream>>>(hb0, asrc[l], adst[l], as_, ad_);
    node_gat_aggr<<<g_node, TB, 0, stream>>>(hb0, as_, ad_, row_start, csr_src,
                                             bias[l], hb1);
    lin = hb1;
    K   = FEAT;
  }

  pool_zero<<<cdiv(N_GRAPHS * FEAT, TB), TB, 0, stream>>>(pooled);
  pool_sum <<<g_nf4, TB, 0, stream>>>(hb1, batch, pooled);
  final_fc <<<N_GRAPHS, TB, 0, stream>>>(pooled, fcW, fcb, out);
}